// WithinImageLoss_29746943492120
// MI455X (gfx1250) — compile-verified
//
#include <hip/hip_runtime.h>
#include <math.h>

#define NPIX 4225
#define KDIM 256
#define INV_TEMP (1.0f / 0.07f)

typedef float v2f __attribute__((ext_vector_type(2)));
typedef float v8f __attribute__((ext_vector_type(8)));

// ---------------------------------------------------------------------------
// Kernel 1: per-class key aggregates v0/v1 (ws[0..255], ws[256..511]),
// class counts N20/N21 (ws[512], ws[513]), and zero the output scalar.
// ---------------------------------------------------------------------------
__global__ __launch_bounds__(256) void prep_kernel(
    const float* __restrict__ x1, const int* __restrict__ x1o,
    float* __restrict__ ws, float* __restrict__ out) {
  __shared__ float r0[256], r1[256], rc[256];
  const int c = blockIdx.x;   // channel 0..255
  const int t = threadIdx.x;
  float s0 = 0.f, s1 = 0.f, cnt1 = 0.f;
  const float* col = x1 + (size_t)c * NPIX;
  for (int i = t; i < NPIX; i += 256) {
    float v = col[i];
    int lb = x1o[i];
    if (lb == 0) s0 += v; else s1 += v;
    if (c == 0 && lb != 0) cnt1 += 1.0f;
  }
  r0[t] = s0; r1[t] = s1; rc[t] = cnt1;
  __syncthreads();
  for (int off = 128; off > 0; off >>= 1) {
    if (t < off) { r0[t] += r0[t + off]; r1[t] += r1[t + off]; rc[t] += rc[t + off]; }
    __syncthreads();
  }
  if (t == 0) {
    ws[c]       = r0[0];   // v0[c] : sum of x1[c,:] over label-0 keys
    ws[256 + c] = r1[0];   // v1[c] : sum over label-1 keys
    if (c == 0) {
      float n21 = rc[0];
      ws[512] = (float)NPIX - n21;  // N20
      ws[513] = n21;                // N21
      out[0] = 0.0f;                // zero accumulator every launch
    }
  }
}

// ---------------------------------------------------------------------------
// Kernel 2: fused GEMM (WMMA f32 16x16x4) + streaming logsumexp + loss reduce.
// Block = 256 threads (8 waves). Block owns 32 query pixels; key loop walks
// 4225 keys in 64-wide tiles (2x4 waves -> 32x64 S tile per iteration).
// ---------------------------------------------------------------------------
__global__ __launch_bounds__(256) void loss_kernel(
    const float* __restrict__ x0, const float* __restrict__ x1,
    const int* __restrict__ x0o, const float* __restrict__ ws,
    float* __restrict__ out) {
  __shared__ float smem[32][65];                 // S tile, padded stride
  __shared__ float redmx[256], redsm[256], reddp[256];

  const int t    = threadIdx.x;
  const int wave = t >> 5;        // 0..7 (wave32)
  const int lane = t & 31;
  const int half = lane >> 4;     // hi/lo half-wave
  const int l16  = lane & 15;
  const int mblk = wave >> 2;     // 0..1 : query 16-row block
  const int nblk = wave & 3;      // 0..3 : key 16-col block

  const int qbase = blockIdx.x * 32;

  // A-fragment query index (clamped; invalid rows masked at the end)
  const int qa   = qbase + mblk * 16 + l16;
  const int qa_c = qa < NPIX ? qa : 0;

  // reduction mapping: 8 threads per query row, 8 key-cols each
  const int row  = t >> 3;        // 0..31
  const int sub  = t & 7;         // 0..7
  const int q    = qbase + row;
  const bool qok = q < NPIX;

  float mx = -INFINITY;           // per-thread running max (over its cols)
  float sm = 0.0f;                // per-thread running sum of exp

  for (int keybase = 0; keybase < NPIX; keybase += 64) {
    const int key   = keybase + nblk * 16 + l16;
    const int key_c = key < NPIX ? key : 0;   // clamped; cols masked below

    v8f acc = {};
    const float* pa = x0 + qa_c;
    const float* pb = x1 + key_c;
#pragma unroll 8
    for (int kk = 0; kk < KDIM; kk += 4) {
      const int k0 = kk + 2 * half;           // K = v + 2*half, v in {0,1}
      v2f a, b;
      a.x = pa[(size_t)(k0 + 0) * NPIX];
      a.y = pa[(size_t)(k0 + 1) * NPIX];
      b.x = pb[(size_t)(k0 + 0) * NPIX];
      b.y = pb[(size_t)(k0 + 1) * NPIX];
      acc = __builtin_amdgcn_wmma_f32_16x16x4_f32(
          /*neg_a=*/false, a, /*neg_b=*/false, b,
          /*c_mod=*/(short)0, acc, /*reuse_a=*/false, /*reuse_b=*/false);
    }
    // C/D layout: VGPR r, lane-half h -> M = r + 8h ; N = lane%16
#pragma unroll
    for (int r = 0; r < 8; ++r)
      smem[mblk * 16 + r + 8 * half][nblk * 16 + l16] = acc[r];
    __syncthreads();

    // streaming logsumexp update over this key tile
#pragma unroll
    for (int i = 0; i < 8; ++i) {
      int col = sub * 8 + i;
      if (keybase + col < NPIX) {
        float v = smem[row][col] * INV_TEMP;
        if (v > mx) { sm = sm * expf(mx - v) + 1.0f; mx = v; }
        else        { sm += expf(v - mx); }
      }
    }
    __syncthreads();
  }

  // partials -> LDS
  redmx[t] = mx;
  redsm[t] = sm;

  // distributed masked-positive dot: dot(x0[:,q], v_class)/TEMP
  float dp = 0.0f;
  int lbl = 0;
  if (qok) {
    lbl = x0o[q];
    const float* vsel = ws + (lbl == 0 ? 0 : 256);
    const float* pq = x0 + q;
    for (int c = sub * 32; c < sub * 32 + 32; ++c)
      dp += pq[(size_t)c * NPIX] * vsel[c];
  }
  reddp[t] = dp;
  __syncthreads();

  if (sub == 0 && qok) {
    float m = redmx[t];
#pragma unroll
    for (int i = 1; i < 8; ++i) m = fmaxf(m, redmx[t + i]);
    float s = 0.0f;
#pragma unroll
    for (int i = 0; i < 8; ++i) s += redsm[t + i] * expf(redmx[t + i] - m);
    float L = logf(s) + m;                       // logsumexp over all keys
    float d = 0.0f;
#pragma unroll
    for (int i = 0; i < 8; ++i) d += reddp[t + i];
    float msum = d * INV_TEMP;
    float Nc = (lbl == 0) ? ws[512] : ws[513];   // == cnt[q] and N2 denom
    float contrib = msum / Nc - L;               // N2*(masked_sum - cnt*L)
    atomicAdd(out, -contrib / (float)NPIX);
  }
}

// ---------------------------------------------------------------------------
extern "C" void kernel_launch(void* const* d_in, const int* in_sizes, int n_in,
                              void* d_out, int out_size, void* d_ws, size_t ws_size,
                              hipStream_t stream) {
  const float* x0  = (const float*)d_in[0];
  const float* x1  = (const float*)d_in[1];
  const int*   x0o = (const int*)d_in[2];
  const int*   x1o = (const int*)d_in[3];
  float* out = (float*)d_out;
  float* ws  = (float*)d_ws;

  prep_kernel<<<KDIM, 256, 0, stream>>>(x1, x1o, ws, out);
  const int nblocks = (NPIX + 31) / 32;  // 133
  loss_kernel<<<nblocks, 256, 0, stream>>>(x0, x1, x0o, ws, out);
}